// AffinityLoss_32220844655139
// MI455X (gfx1250) — compile-verified
//
#include <hip/hip_runtime.h>

typedef float v2f __attribute__((ext_vector_type(2)));
typedef float v8f __attribute__((ext_vector_type(8)));

#define H    512
#define W    512
#define OH   510
#define OW   510
#define NB   4
#define TILE 16
#define GXB  32   // ceil(510/16)
#define GYB  32
#define NPART (GXB * GYB * NB)
// n * 9 * 9 * L = 4 * 81 * 510 * 510
#define TOTAL_TERMS 84272400.0f
#define LN2F        0.69314718055994530942f
// clamp at -100 in ln-domain == -100/ln2 in log2-domain
#define CLAMP_LOG2  (-144.26950408889634f)

// Lane-XOR add via ds_swizzle (group-of-32 mode: offset = (xor<<10)|0x1f),
// no bounds-check cndmask overhead like generic __shfl_xor.
#define SWZ_XOR_ADD(s, mask)                                                   \
    (s) += __int_as_float(__builtin_amdgcn_ds_swizzle(                         \
               __float_as_int(s), (((mask) << 10) | 0x1f)))

// Exact sum of the 32 lane values of this wave: one V_WMMA_F32_16X16X4_F32
// with A = ones column-sums the B operand (each lane contributes v in B.x,
// 0 in B.y); folding the 16 column sums with 4 xor-swizzles then yields the
// sum of EVERY element fed into B, independent of the exact B VGPR striping.
// EXEC must be all-ones for the wave when this is called.
__device__ __forceinline__ float wave_reduce_wmma(float v) {
    v2f a; a[0] = 1.0f; a[1] = 1.0f;   // A = ones (layout-invariant)
    v2f b; b[0] = v;    b[1] = 0.0f;
    v8f c = {};
    c = __builtin_amdgcn_wmma_f32_16x16x4_f32(
            /*neg_a=*/false, a, /*neg_b=*/false, b,
            /*c_mod=*/(short)0, c, /*reuse_a=*/false, /*reuse_b=*/false);
    float s = c[0];                    // lane n: column sum (rows identical)
    SWZ_XOR_ADD(s, 8);
    SWZ_XOR_ADD(s, 4);
    SWZ_XOR_ADD(s, 2);
    SWZ_XOR_ADD(s, 1);
    return s;                          // every lane: wave total
}

__global__ __launch_bounds__(256)
void aff_partial_kernel(const float* __restrict__ logits,
                        const float* __restrict__ labels,
                        float* __restrict__ part) {
    __shared__ float sp[18][20];   // sigmoid(logits) tile + halo (padded stride)
    __shared__ float sl[18][20];   // labels tile + halo
    __shared__ float wsum[8];

    const int tx  = threadIdx.x, ty = threadIdx.y;
    const int tid = ty * 16 + tx;
    const int n   = blockIdx.z;
    const int ox0 = blockIdx.x * TILE;
    const int oy0 = blockIdx.y * TILE;

    const float* lg = logits + (size_t)n * H * W;
    const float* lb = labels + (size_t)n * H * W;

    // Cooperative halo load: 18x18 values, sigmoid computed once per pixel.
    for (int idx = tid; idx < 18 * 18; idx += 256) {
        const int r  = idx / 18, cc = idx % 18;
        const int gy = oy0 + r,  gx = ox0 + cc;
        float pv = 0.5f;   // safe pad: keeps logs finite for masked threads
        float lv = 0.0f;
        if (gy < H && gx < W) {
            const float x = lg[gy * W + gx];
            pv = __builtin_amdgcn_rcpf(1.0f + __expf(-x));  // v_exp + v_rcp
            lv = lb[gy * W + gx];
        }
        sp[r][cc] = pv;
        sl[r][cc] = lv;
    }
    __syncthreads();

    // 3x3 neighborhood into registers; q = 1-p precomputed (9 subs).
    float p[9], q[9], l[9];
#pragma unroll
    for (int i = 0; i < 3; ++i)
#pragma unroll
        for (int j = 0; j < 3; ++j) {
            const float pv = sp[ty + i][tx + j];
            p[i * 3 + j] = pv;
            q[i * 3 + j] = 1.0f - pv;
            l[i * 3 + j] = sl[ty + i][tx + j];
        }

    // 45 unique pairs (symmetry): diagonal terms are always the log(aff)
    // branch (la==la); off-diagonal terms weighted 2x via one final FMA.
    // Accumulate in log2 domain (raw v_log_f32), convert with one mul.
    float accDiag = 0.0f, accOff = 0.0f;
#pragma unroll
    for (int a2 = 0; a2 < 9; ++a2) {
        // diagonal: aff = pa^2 + qa^2
        {
            const float pa = p[a2], qa = q[a2];
            const float aff = fmaf(pa, pa, qa * qa);
            accDiag += fmaxf(__builtin_amdgcn_logf(aff), CLAMP_LOG2);
        }
#pragma unroll
        for (int b2 = a2 + 1; b2 < 9; ++b2) {
            const float pa = p[a2], pb = p[b2];
            const float aff  = fmaf(pa, pb, q[a2] * q[b2]); // (1-pa)(1-pb)+pa*pb
            const float offv = 1.0f - aff;                  // matches reference
            const float x    = (l[a2] == l[b2]) ? aff : offv;
            accOff += fmaxf(__builtin_amdgcn_logf(x), CLAMP_LOG2);
        }
    }
    float acc = fmaf(2.0f, accOff, accDiag) * LN2F;

    // Mask out-of-range output locations multiplicatively (no divergence,
    // so EXEC stays all-ones for the WMMA reduction below).
    const float valid = ((oy0 + ty) < OH && (ox0 + tx) < OW) ? 1.0f : 0.0f;
    acc *= valid;

    // Per-wave exact reduction via WMMA, then cross-wave via LDS.
    const float wtot = wave_reduce_wmma(acc);
    const int lane = tid & 31, wid = tid >> 5;
    if (lane == 0) wsum[wid] = wtot;
    __syncthreads();
    if (wid == 0) {
        float v = (lane < 8) ? wsum[lane] : 0.0f;
        SWZ_XOR_ADD(v, 4);
        SWZ_XOR_ADD(v, 2);
        SWZ_XOR_ADD(v, 1);
        if (lane == 0) {
            const int bid = (blockIdx.z * GYB + blockIdx.y) * GXB + blockIdx.x;
            part[bid] = v;
        }
    }
}

__global__ __launch_bounds__(256)
void aff_final_kernel(const float* __restrict__ part, float* __restrict__ out) {
    __shared__ float sm[256];
    const int tid = threadIdx.x;
    float s = 0.0f;
    for (int i = tid; i < NPART; i += 256) s += part[i];
    sm[tid] = s;
    __syncthreads();
    if (tid < 32) {   // wave 0 only; all 32 lanes active -> EXEC all-ones
        float t = 0.0f;
#pragma unroll
        for (int k = 0; k < 8; ++k) t += sm[tid * 8 + k];
        const float tot = wave_reduce_wmma(t);
        if (tid == 0) out[0] = -tot * (1.0f / TOTAL_TERMS);
    }
}

extern "C" void kernel_launch(void* const* d_in, const int* in_sizes, int n_in,
                              void* d_out, int out_size, void* d_ws, size_t ws_size,
                              hipStream_t stream) {
    const float* logits = (const float*)d_in[0];
    const float* labels = (const float*)d_in[1];
    float* part = (float*)d_ws;              // NPART floats = 16 KB scratch

    dim3 block(16, 16, 1);
    dim3 grid(GXB, GYB, NB);
    aff_partial_kernel<<<grid, block, 0, stream>>>(logits, labels, part);
    aff_final_kernel<<<1, 256, 0, stream>>>(part, (float*)d_out);
}